// MLPLayer_42013370089734
// MI455X (gfx1250) — compile-verified
//
#include <hip/hip_runtime.h>
#include <math.h>

typedef float v2f __attribute__((ext_vector_type(2)));
typedef float v8f __attribute__((ext_vector_type(8)));
typedef int   v4i __attribute__((vector_size(16)));
typedef __attribute__((address_space(1))) v4i gv4i;   // global
typedef __attribute__((address_space(3))) v4i lv4i;   // LDS

#define LN_EPS 1e-5f
#define TILES_PER_WAVE 2

// LDS float offsets
#define W1P_OFF 0        // 16*512 = 8192 floats (W1 zero-padded 8->16 rows)
#define W2_OFF  8192     // 512*8  = 4096
#define B2_OFF  12288    // 512
#define G_OFF   12800    // 512
#define BE_OFF  13312    // 512
#define B1_OFF  13824    // 8 (+pad)
#define H_OFF   13840    // 8 waves * 128
#define XS_OFF  14864    // 8 waves * 16*512 = 65536
#define SMEM_FLOATS 80400  // 321600 B < 320 KB WGP LDS

#if __has_builtin(__builtin_amdgcn_global_load_async_to_lds_b128)
#define ASYNC_LDS 1
#else
#define ASYNC_LDS 0
#endif

__device__ __forceinline__ void copy16(const float4* g, float4* l) {
#if ASYNC_LDS
    // global src (AS1), LDS dst (AS3); flat LDS addr low 32 bits == LDS offset
    __builtin_amdgcn_global_load_async_to_lds_b128(
        (gv4i*)(size_t)g, (lv4i*)(unsigned)(size_t)l, 0, 0);
#else
    *l = *g;
#endif
}

__device__ __forceinline__ void wait_async() {
#if ASYNC_LDS
#if __has_builtin(__builtin_amdgcn_s_wait_asynccnt)
    __builtin_amdgcn_s_wait_asynccnt(0);
#else
    asm volatile("s_wait_asynccnt 0x0" ::: "memory");
#endif
#endif
}

__device__ __forceinline__ void drain_lds_reads() {
#if ASYNC_LDS
    // WAR guard: async LDS writes are unordered vs DS reads of previous tile
    asm volatile("s_wait_dscnt 0x0" ::: "memory");
#endif
}

// exact-GELU via Abramowitz–Stegun 7.1.26 erf (|err| < 1.5e-7), branch-free:
// uses v_rcp_f32 + v_exp_f32 (TRANS ops co-execute with VALU)
__device__ __forceinline__ float gelu_fast(float v) {
    float ax = fabsf(v) * 0.70710678118654752440f;          // |v|/sqrt(2)
    float t  = __builtin_amdgcn_rcpf(fmaf(0.3275911f, ax, 1.0f));
    float p  = t * fmaf(t, fmaf(t, fmaf(t, fmaf(t, 1.061405429f, -1.453152027f),
                       1.421413741f), -0.284496736f), 0.254829592f);
    float e  = __expf(-ax * ax);
    float er = copysignf(fmaf(-p, e, 1.0f), v);             // erf(v/sqrt2)
    return fmaf(0.5f * v, er, 0.5f * v);                    // 0.5*v*(1+erf)
}

extern "C" __global__ __launch_bounds__(256, 1)
void mlp_fused_wmma(const float* __restrict__ x,
                    const float* __restrict__ w1,
                    const float* __restrict__ b1,
                    const float* __restrict__ w2,
                    const float* __restrict__ b2,
                    const float* __restrict__ gamma,
                    const float* __restrict__ beta,
                    float* __restrict__ out)
{
    extern __shared__ float sm[];
    const int tid = threadIdx.x;

    // ---------------- stage weights into LDS (once per block) ----------------
    {
        const float4* w1g = (const float4*)w1;
        float4* w14 = (float4*)(sm + W1P_OFF);
        const float4 z4 = {0.f, 0.f, 0.f, 0.f};
        for (int i = tid; i < 16 * 512 / 4; i += 256) {
            if (i < 8 * 512 / 4) copy16(w1g + i, w14 + i);
            else                 w14[i] = z4;                 // N padding rows 8..15
        }
        const float4* w2g = (const float4*)w2;
        float4* w24 = (float4*)(sm + W2_OFF);
        for (int i = tid; i < 512 * 8 / 4; i += 256) copy16(w2g + i, w24 + i);
        if (tid < 128) {
            copy16((const float4*)b2 + tid,    (float4*)(sm + B2_OFF) + tid);
            copy16((const float4*)gamma + tid, (float4*)(sm + G_OFF) + tid);
            copy16((const float4*)beta + tid,  (float4*)(sm + BE_OFF) + tid);
        }
        if (tid < 8) sm[B1_OFF + tid] = b1[tid];
        wait_async();
    }
    __syncthreads();

    const int wave = tid >> 5;        // 0..7 : each wave owns a 16-token M-tile stream
    const int lane = tid & 31;
    const int half = lane >> 4;       // which K-pair this lane carries in A/B
    const int tl   = lane & 15;       // M index (A) and N index (B/C)

    float* xs = sm + XS_OFF + wave * (16 * 512);   // this wave's x slab (32 KB)
    float* hb = sm + H_OFF  + wave * 128;          // this wave's hidden tile [16][8]

    for (int t = 0; t < TILES_PER_WAVE; ++t) {
        const size_t tile = ((size_t)blockIdx.x * 8 + wave) * TILES_PER_WAVE + t;
        const float* xg = x + tile * (16 * 512);

        // -------- wave-local async stage of 16x512 x tile (no barrier) --------
        {
            const float4* xg4 = (const float4*)xg;
            float4* xs4 = (float4*)xs;
            #pragma unroll 8
            for (int i = lane; i < 16 * 512 / 4; i += 32) copy16(xg4 + i, xs4 + i);
            wait_async();
        }

        // -------- LayerNorm stats (wave32: per-lane partials + shfl_xor 16) --------
        float s = 0.f, s2 = 0.f;
        {
            const float4* xr = (const float4*)(xs + tl * 512 + half * 256);
            #pragma unroll
            for (int j = 0; j < 64; ++j) {
                float4 v = xr[j];
                s  += v.x + v.y + v.z + v.w;
                s2 += v.x*v.x + v.y*v.y + v.z*v.z + v.w*v.w;
            }
        }
        s  += __shfl_xor(s, 16, 32);
        s2 += __shfl_xor(s2, 16, 32);
        const float mu   = s * (1.f / 512.f);
        const float var  = s2 * (1.f / 512.f) - mu * mu;
        const float rstd = rsqrtf(var + LN_EPS);

        // -------- GEMM1 via WMMA: C[16x16] = LN(x)[16x512] @ W1p^T --------
        v8f acc = {0.f,0.f,0.f,0.f,0.f,0.f,0.f,0.f};
        {
            const int dk = 2 * half;
            #pragma unroll 4
            for (int k = 0; k < 128; ++k) {
                const int d = 4 * k + dk;
                v2f xp = *(const v2f*)(xs + tl * 512 + d);
                v2f gp = *(const v2f*)(sm + G_OFF + d);
                v2f bp = *(const v2f*)(sm + BE_OFF + d);
                v2f a;
                a.x = fmaf((xp.x - mu) * rstd, gp.x, bp.x);
                a.y = fmaf((xp.y - mu) * rstd, gp.y, bp.y);
                v2f bm = *(const v2f*)(sm + W1P_OFF + tl * 512 + d);
                acc = __builtin_amdgcn_wmma_f32_16x16x4_f32(
                    false, a, false, bm, (short)0, acc, false, false);
            }
        }

        // -------- hidden: H = gelu(C + b1), stored in GEMM2 A-layout --------
        if (tl < 8) {
            const float b1n = sm[B1_OFF + tl];
            #pragma unroll
            for (int r = 0; r < 8; ++r)
                hb[(r + 8 * half) * 8 + tl] = gelu_fast(acc[r] + b1n);
        }

        // -------- GEMM2 via WMMA + fused epilogue --------
        const int kk = 2 * half;
        v2f a0 = *(const v2f*)(hb + tl * 8 + kk);       // K 0..3
        v2f a1 = *(const v2f*)(hb + tl * 8 + 4 + kk);   // K 4..7

        float* outg = out + tile * (16 * 512);
        #pragma unroll 2
        for (int nt = 0; nt < 32; ++nt) {
            const int n = nt * 16 + tl;
            v2f bA = *(const v2f*)(sm + W2_OFF + n * 8 + kk);
            v2f bB = *(const v2f*)(sm + W2_OFF + n * 8 + 4 + kk);
            v8f c2 = {0.f,0.f,0.f,0.f,0.f,0.f,0.f,0.f};
            c2 = __builtin_amdgcn_wmma_f32_16x16x4_f32(false, a0, false, bA, (short)0, c2, false, false);
            c2 = __builtin_amdgcn_wmma_f32_16x16x4_f32(false, a1, false, bB, (short)0, c2, false, false);

            const float b2n = sm[B2_OFF + n];
            #pragma unroll
            for (int r = 0; r < 8; ++r) {
                const int tloc = r + 8 * half;
                float xv = xs[tloc * 512 + n];
                outg[(size_t)tloc * 512 + n] = gelu_fast(c2[r] + b2n) + xv;
            }
        }
        drain_lds_reads();   // safe to overwrite slab with next tile's async loads
    }
}

extern "C" void kernel_launch(void* const* d_in, const int* in_sizes, int n_in,
                              void* d_out, int out_size, void* d_ws, size_t ws_size,
                              hipStream_t stream) {
    const float* x     = (const float*)d_in[0];
    const float* w1    = (const float*)d_in[1];
    const float* b1    = (const float*)d_in[2];
    const float* w2    = (const float*)d_in[3];
    const float* b2    = (const float*)d_in[4];
    const float* gamma = (const float*)d_in[5];
    const float* beta  = (const float*)d_in[6];
    float* out = (float*)d_out;

    const int tokens = in_sizes[0] / 512;                    // 131072
    const int blocks = tokens / (16 * 8 * TILES_PER_WAVE);   // 512
    const size_t smem = (size_t)SMEM_FLOATS * sizeof(float);

    hipLaunchKernelGGL(mlp_fused_wmma, dim3(blocks), dim3(256), smem, stream,
                       x, w1, b1, w2, b2, gamma, beta, out);
}